// CRF_3882650436048
// MI455X (gfx1250) — compile-verified
//
#include <hip/hip_runtime.h>
#include <stdint.h>

#define Bn 512
#define Sn 1024
#define Tn 50
#define STARTn 48
#define STOPn 49

#define CHn 64                      // timesteps per feats chunk
#define NCH (Sn / CHn)              // 16 chunks
#define CHUNK_FLOATS (CHn * Tn)     // 3200 floats
#define CHUNK_BYTES  (CHUNK_FLOATS * 4)  // 12800 B
#define SWEEPS (CHUNK_BYTES / (64 * 8))  // 25 async b64 instructions per wave per chunk

#define NEG_INF (-3.402823466e38f)

__device__ __forceinline__ void async_copy_b64(unsigned lds_off, unsigned long long gaddr) {
    // CDNA5 async global -> LDS DMA (tracked with ASYNCcnt)
    asm volatile("global_load_async_to_lds_b64 %0, %1, off"
                 :: "v"(lds_off), "v"(gaddr) : "memory");
}
__device__ __forceinline__ void wait_async_le_sweeps() {
    // ASYNCcnt <= 25: everything except the most recent chunk's loads is complete
    asm volatile("s_wait_asynccnt 0x19" ::: "memory");
}

__global__ __launch_bounds__(64) void viterbi_crf_kernel(
    const float* __restrict__ feats,
    const unsigned char* __restrict__ mask,
    const float* __restrict__ trans,
    int* __restrict__ out)
{
    __shared__ float chunkbuf[2][CHUNK_FLOATS];        // 25,600 B
    __shared__ float partbuf[2][52];                   // double-buffered partition
    __shared__ float lastpart[52];
    __shared__ int   decode_sh[Sn];                    // 4,096 B
    __shared__ unsigned char bp_sh[(Sn - 1) * Tn];     // 51,150 B backpointers (t=1..S-1)
    __shared__ int   len_sh;

    const int b   = blockIdx.x;
    const int tid = threadIdx.x;
    const int k   = (tid < Tn) ? tid : (Tn - 1);
    const bool active = (tid < Tn);

    const float* frow         = feats + (size_t)b * Sn * Tn;
    const unsigned char* mrow = mask  + (size_t)b * Sn;

    // transitions column k resident in registers (50 VGPRs)
    float tr[Tn];
#pragma unroll
    for (int j = 0; j < Tn; ++j) tr[j] = trans[j * Tn + k];

    // sequence length = sum(mask row)
    if (tid == 0) len_sh = 0;
    __syncthreads();
    {
        int psum = 0;
        for (int i = tid; i < Sn; i += 64) psum += (int)mrow[i];
        atomicAdd(&len_sh, psum);
    }

    // ---- prologue: async-prefetch chunks 0 and 1 ----
    const unsigned lbase0 = (unsigned)(uintptr_t)(&chunkbuf[0][0]);
    const unsigned lbase1 = (unsigned)(uintptr_t)(&chunkbuf[1][0]);
    const unsigned long long gbase = (unsigned long long)(uintptr_t)frow;
    {
        const unsigned lane_off = (unsigned)tid * 8u;
#pragma unroll
        for (int s = 0; s < SWEEPS; ++s) {
            unsigned off = (unsigned)s * 512u + lane_off;
            async_copy_b64(lbase0 + off, gbase + off);
        }
#pragma unroll
        for (int s = 0; s < SWEEPS; ++s) {
            unsigned off = (unsigned)s * 512u + lane_off;
            async_copy_b64(lbase1 + off, gbase + CHUNK_BYTES + off);
        }
    }
    wait_async_le_sweeps();          // chunk 0 resident
    __syncthreads();
    const int length = len_sh;

    // ---- forward Viterbi ----
    int cur = 0;
    for (int c = 0; c < NCH; ++c) {
        const float* fb = &chunkbuf[c & 1][0];
        for (int tt = 0; tt < CHn; ++tt) {
            const int t = c * CHn + tt;
            const float f = fb[tt * Tn + k];
            float np;
            if (t == 0) {
                np = f + tr[STARTn];           // partition0 = feats[0] + trans[START]
            } else {
                // cur[j,k] = (f[k] + trans[j,k]) + part[j]  (exact FP order of reference)
                // 4 independent chains over j-subranges for ILP; merge prefers lower
                // range on ties -> exact first-occurrence argmax semantics.
                float b0 = NEG_INF, b1 = NEG_INF, b2 = NEG_INF, b3 = NEG_INF;
                int   a0 = 0, a1 = 0, a2 = 0, a3 = 0;
                const float* pc = &partbuf[cur][0];
#pragma unroll
                for (int j = 0; j < 13; ++j) {
                    float v = (f + tr[j]) + pc[j];
                    bool g = v > b0; a0 = g ? j : a0; b0 = g ? v : b0;
                }
#pragma unroll
                for (int j = 13; j < 26; ++j) {
                    float v = (f + tr[j]) + pc[j];
                    bool g = v > b1; a1 = g ? j : a1; b1 = g ? v : b1;
                }
#pragma unroll
                for (int j = 26; j < 38; ++j) {
                    float v = (f + tr[j]) + pc[j];
                    bool g = v > b2; a2 = g ? j : a2; b2 = g ? v : b2;
                }
#pragma unroll
                for (int j = 38; j < Tn; ++j) {
                    float v = (f + tr[j]) + pc[j];
                    bool g = v > b3; a3 = g ? j : a3; b3 = g ? v : b3;
                }
                if (b1 > b0) { b0 = b1; a0 = a1; }   // tie -> lower range wins
                if (b3 > b2) { b2 = b3; a2 = a3; }
                if (b2 > b0) { b0 = b2; a0 = a2; }
                np = b0;
                const unsigned char mb = mrow[t];
                if (active)
                    bp_sh[(t - 1) * Tn + tid] = mb ? (unsigned char)a0 : (unsigned char)0;
            }
            if (active) {
                partbuf[cur ^ 1][tid] = np;
                if (t == length - 1) lastpart[tid] = np;   // snapshot partition_history[len-1]
            }
            __syncthreads();
            cur ^= 1;
        }
        // ---- chunk boundary: prefetch chunk (c+2) (wrapped) into the buffer just freed ----
        {
            const int cc = (c + 2) & (NCH - 1);
            const unsigned lb = (c & 1) ? lbase1 : lbase0;
            const unsigned long long gg = gbase + (unsigned long long)cc * CHUNK_BYTES;
            const unsigned lane_off = (unsigned)tid * 8u;
#pragma unroll
            for (int s = 0; s < SWEEPS; ++s) {
                unsigned off = (unsigned)s * 512u + lane_off;
                async_copy_b64(lb + off, gg + off);
            }
        }
        wait_async_le_sweeps();      // chunk c+1 fully resident for this wave
        __syncthreads();             // ... and for all waves
    }

    // ---- epilogue: final transition argmax + in-LDS backtrace (thread 0) ----
    if (tid == 0) {
        float best = NEG_INF;
        int p0 = 0;
        for (int j = 0; j < Tn; ++j) {
            float v = lastpart[j] + trans[j * Tn + STOPn];  // part + trans (reference order)
            if (v > best) { best = v; p0 = j; }
        }
        int ptr = p0;
        decode_sh[Sn - 1] = p0;
        for (int i = Sn - 2; i >= 0; --i) {
            ptr = (i == length - 1) ? p0 : (int)bp_sh[i * Tn + ptr];
            decode_sh[i] = ptr;
        }
    }
    __syncthreads();

    int* orow = out + (size_t)b * Sn;
    for (int i = tid; i < Sn; i += 64) orow[i] = decode_sh[i];
}

extern "C" void kernel_launch(void* const* d_in, const int* in_sizes, int n_in,
                              void* d_out, int out_size, void* d_ws, size_t ws_size,
                              hipStream_t stream) {
    (void)in_sizes; (void)n_in; (void)d_ws; (void)ws_size; (void)out_size;
    const float*         feats = (const float*)d_in[0];
    const unsigned char* mask  = (const unsigned char*)d_in[1];
    const float*         trans = (const float*)d_in[2];
    int*                 out   = (int*)d_out;
    viterbi_crf_kernel<<<Bn, 64, 0, stream>>>(feats, mask, trans, out);
}